// MultiHeadGATLayer_80444737454869
// MI455X (gfx1250) — compile-verified
//
#include <hip/hip_runtime.h>
#include <math.h>

// MultiHeadGAT for MI455X (gfx1250, wave32).
//
// Pipeline (all on `stream`, graph-capture safe):
//   0. prep_c3:      c3[h][16] = W_edge[h] @ a3[h]          (folds edge_proj away)
//   1. gemm xw:      xw[N][256] = x @ concat_h(W[h])        (v_wmma_f32_16x16x4_f32)
//   2. node_scores:  s1[n][h] = xw[n,h]·a1[h], s2 likewise  (wave-per-(n,h) dot)
//   3. CSR build:    degree count -> prefix scan -> scatter (only ~3.2M int atomics)
//   4. edge_logits:  e[h,edge] = leakyrelu(s1[dst]+s2[src]+edge_attr·c3[h])
//   5. aggregate:    per dst node: segment max/sum softmax + weighted gather of
//                    xw[src] rows (coalesced 1KB loads), no atomics
//   6. gemm proj:    out = ELU(out_h @ proj_w + b)          (WMMA + epilogue)

typedef __attribute__((ext_vector_type(2))) float v2f;
typedef __attribute__((ext_vector_type(8))) float v8f;

#define HID 64
#define NHEAD 4
#define DIM 256           // IN_DIM == H*HID == OUT_DIM == 256
#define NEG_SLOPE 0.2f

// ---------------------------------------------------------------- c3 = W_edge @ a3
__global__ void prep_c3(const float* __restrict__ W_edge, const float* __restrict__ att,
                        float* __restrict__ c3) {
    int t = threadIdx.x;
    if (t >= NHEAD * 16) return;
    int h = t >> 4, d = t & 15;
    const float* w  = W_edge + (size_t)(h * 16 + d) * HID;
    const float* a3 = att + h * (3 * HID) + 2 * HID;
    float s = 0.f;
    #pragma unroll
    for (int o = 0; o < HID; ++o) s += w[o] * a3[o];
    c3[t] = s;
}

// ------------------------------------------------- WMMA GEMM: (M x 256) @ (256 x 256)
// MODE 0: B[i][j] = W[(j>>6)][i][j&63]  (concat of per-head W), plain store.
// MODE 1: B = proj_w row-major, epilogue = +bias then ELU.
// One wave per 16x16 output tile; 8 waves (8 col tiles) per 256-thread block.
template <int MODE>
__global__ void gemm256_wmma(const float* __restrict__ A, const float* __restrict__ B,
                             const float* __restrict__ bias, float* __restrict__ D) {
    const int lane   = threadIdx.x & 31;
    const int wave   = threadIdx.x >> 5;
    const int tile_n = blockIdx.x * 8 + wave;   // 0..15
    const int tile_m = blockIdx.y;              // 0..N/16-1

    const int col  = tile_n * 16 + (lane & 15); // output / B column
    const int arow = tile_m * 16 + (lane & 15); // A row for this lane
    const int kb   = (lane >> 4) << 1;          // lanes 0-15 -> K{0,1}; 16-31 -> K{2,3}

    v8f acc = {};
    for (int k = 0; k < DIM; k += 4) {
        const int k0 = k + kb;
        // A fragment: 16x4 f32, two consecutive K per lane (aligned float2)
        v2f a = *(const v2f*)(A + (size_t)arow * DIM + k0);
        // B fragment: 4x16 f32, rows K=k0,k0+1 at column `col`
        v2f b;
        if (MODE == 0) {
            const float* wp = B + (size_t)(col >> 6) * (DIM * HID) + (col & 63);
            b.x = wp[(size_t)k0 * HID];
            b.y = wp[(size_t)(k0 + 1) * HID];
        } else {
            b.x = B[(size_t)k0 * DIM + col];
            b.y = B[(size_t)(k0 + 1) * DIM + col];
        }
        acc = __builtin_amdgcn_wmma_f32_16x16x4_f32(
            /*neg_a=*/false, a, /*neg_b=*/false, b,
            /*c_mod=*/(short)0, acc, /*reuse_a=*/false, /*reuse_b=*/false);
    }

    // C/D layout: VGPR v -> row v (lanes 0-15) / row v+8 (lanes 16-31)
    const int rbase = tile_m * 16 + ((lane >> 4) << 3);
    #pragma unroll
    for (int v = 0; v < 8; ++v) {
        float val = acc[v];
        if (MODE == 1) {
            val += bias[col];
            val = (val > 0.f) ? val : expm1f(val);   // ELU(alpha=1)
        }
        D[(size_t)(rbase + v) * DIM + col] = val;
    }
}

// ----------------------------------------- s1[n*4+h], s2[n*4+h]: wave per (node,head)
__global__ void node_scores(const float* __restrict__ xw, const float* __restrict__ att,
                            float* __restrict__ s1, float* __restrict__ s2, int N) {
    int gw   = (int)((blockIdx.x * (size_t)blockDim.x + threadIdx.x) >> 5);
    int lane = threadIdx.x & 31;
    if (gw >= N * NHEAD) return;
    int n = gw >> 2, h = gw & 3;
    const float* row = xw + (size_t)n * DIM + h * HID;
    const float* a   = att + h * (3 * HID);
    float x0 = row[lane], x1 = row[lane + 32];
    float p1 = x0 * a[lane]      + x1 * a[lane + 32];
    float p2 = x0 * a[HID + lane] + x1 * a[HID + lane + 32];
    #pragma unroll
    for (int off = 16; off; off >>= 1) {
        p1 += __shfl_xor(p1, off, 32);
        p2 += __shfl_xor(p2, off, 32);
    }
    if (lane == 0) { s1[n * 4 + h] = p1; s2[n * 4 + h] = p2; }
}

// ----------------------------------------------------------------------- CSR build
__global__ void count_deg(const int* __restrict__ dst, int* __restrict__ deg, int E) {
    int e = blockIdx.x * blockDim.x + threadIdx.x;
    if (e < E) atomicAdd(&deg[dst[e]], 1);
}

__global__ void scan_offsets(const int* __restrict__ deg, int* __restrict__ offs, int n) {
    __shared__ int buf[1024];
    __shared__ int carry;
    if (threadIdx.x == 0) { carry = 0; offs[0] = 0; }
    __syncthreads();
    for (int base = 0; base < n; base += 1024) {
        int i = base + (int)threadIdx.x;
        int v = (i < n) ? deg[i] : 0;
        buf[threadIdx.x] = v;
        __syncthreads();
        for (int off = 1; off < 1024; off <<= 1) {
            int t = (threadIdx.x >= (unsigned)off) ? buf[threadIdx.x - off] : 0;
            __syncthreads();
            buf[threadIdx.x] += t;
            __syncthreads();
        }
        int incl = buf[threadIdx.x];
        if (i < n) offs[i + 1] = carry + incl;     // reads carry ...
        __syncthreads();
        if (threadIdx.x == 1023) carry += incl;    // ... before this update
        __syncthreads();
    }
}

__global__ void fill_csr(const int* __restrict__ dst, const int* __restrict__ offs,
                         int* __restrict__ woff, int* __restrict__ csr, int E) {
    int e = blockIdx.x * blockDim.x + threadIdx.x;
    if (e < E) {
        int d = dst[e];
        csr[offs[d] + atomicAdd(&woff[d], 1)] = e;
    }
}

// ------------------------------------------------------- per-edge attention logits
__global__ void edge_logits(const int* __restrict__ src, const int* __restrict__ dst,
                            const float* __restrict__ eattr, const float* __restrict__ c3,
                            const float* __restrict__ s1, const float* __restrict__ s2,
                            float* __restrict__ elog, int E) {
    __shared__ float c3s[NHEAD * 16];
    if (threadIdx.x < NHEAD * 16) c3s[threadIdx.x] = c3[threadIdx.x];
    __syncthreads();
    int e = blockIdx.x * blockDim.x + threadIdx.x;
    if (e >= E) return;

    float ea[16];
    const float4* ep = (const float4*)(eattr) + (size_t)e * 4;
    #pragma unroll
    for (int q = 0; q < 4; ++q) {
        float4 v = ep[q];
        ea[4 * q + 0] = v.x; ea[4 * q + 1] = v.y; ea[4 * q + 2] = v.z; ea[4 * q + 3] = v.w;
    }
    int d = dst[e], s = src[e];
    float4 s1v = ((const float4*)s1)[d];
    float4 s2v = ((const float4*)s2)[s];
    float base[NHEAD] = {s1v.x + s2v.x, s1v.y + s2v.y, s1v.z + s2v.z, s1v.w + s2v.w};

    float4 out;
    float* op = (float*)&out;
    #pragma unroll
    for (int h = 0; h < NHEAD; ++h) {
        float e3 = 0.f;
        #pragma unroll
        for (int q = 0; q < 16; ++q) e3 += ea[q] * c3s[h * 16 + q];
        float v = base[h] + e3;
        op[h] = (v >= 0.f) ? v : NEG_SLOPE * v;    // LeakyReLU
    }
    ((float4*)elog)[e] = out;                      // elog[e*4 + h]
}

// -------------------- per-dst-node softmax + weighted aggregation (block per node)
__global__ void aggregate(const int* __restrict__ offs, const int* __restrict__ csr,
                          const int* __restrict__ src, const float* __restrict__ elog,
                          const float* __restrict__ xw, float* __restrict__ outh) {
    const int n = blockIdx.x;
    const int t = threadIdx.x;        // output channel (h*64 + o)
    const int h = t >> 6;
    const int beg = offs[n], end = offs[n + 1];

    float m = -INFINITY;
    for (int k = beg; k < end; ++k)
        m = fmaxf(m, elog[(size_t)csr[k] * 4 + h]);
    if (!isfinite(m)) m = 0.f;        // reference: non-finite segment max -> 0

    float s = 0.f, acc = 0.f;
    for (int k = beg; k < end; ++k) {
        int e = csr[k];
        float w = __expf(elog[(size_t)e * 4 + h] - m);
        s += w;
        acc += w * xw[(size_t)src[e] * DIM + t];   // coalesced 1KB row gather
    }
    outh[(size_t)n * DIM + t] = acc / (s + 1e-16f);
}

// ---------------------------------------------------------------------------------
extern "C" void kernel_launch(void* const* d_in, const int* in_sizes, int n_in,
                              void* d_out, int out_size, void* d_ws, size_t ws_size,
                              hipStream_t stream) {
    const float* x      = (const float*)d_in[0];
    const int*   ei     = (const int*)d_in[1];    // int32 (JAX default, no x64)
    const float* eattr  = (const float*)d_in[2];
    const float* W      = (const float*)d_in[3];
    const float* W_edge = (const float*)d_in[4];
    const float* att    = (const float*)d_in[5];
    const float* proj_w = (const float*)d_in[6];
    const float* proj_b = (const float*)d_in[7];
    float* out = (float*)d_out;

    const int N = in_sizes[0] / DIM;   // 100000 (divisible by 16)
    const int E = in_sizes[2] / 16;    // 1600000
    const int* src = ei;
    const int* dst = ei + E;

    // carve workspace (256B-aligned chunks), ~241 MB total
    char* p = (char*)d_ws;
    auto carve = [&](size_t bytes) -> void* {
        void* q = (void*)p;
        p += (bytes + 255) & ~(size_t)255;
        return q;
    };
    float* xw   = (float*)carve((size_t)N * DIM * 4);
    float* outh = (float*)carve((size_t)N * DIM * 4);
    float* elog = (float*)carve((size_t)E * NHEAD * 4);
    float* s1   = (float*)carve((size_t)N * NHEAD * 4);
    float* s2   = (float*)carve((size_t)N * NHEAD * 4);
    float* c3   = (float*)carve(NHEAD * 16 * 4);
    int*   deg  = (int*)carve((size_t)N * 4);
    int*   offs = (int*)carve((size_t)(N + 1) * 4);
    int*   woff = (int*)carve((size_t)N * 4);
    int*   csr  = (int*)carve((size_t)E * 4);

    hipMemsetAsync(deg,  0, (size_t)N * 4, stream);
    hipMemsetAsync(woff, 0, (size_t)N * 4, stream);

    prep_c3<<<1, 64, 0, stream>>>(W_edge, att, c3);

    // xw = x @ concat_h(W[h])   (WMMA f32)
    gemm256_wmma<0><<<dim3(2, N / 16), 256, 0, stream>>>(x, W, nullptr, xw);

    node_scores<<<(N * NHEAD * 32 + 255) / 256, 256, 0, stream>>>(xw, att, s1, s2, N);

    count_deg<<<(E + 255) / 256, 256, 0, stream>>>(dst, deg, E);
    scan_offsets<<<1, 1024, 0, stream>>>(deg, offs, N);
    fill_csr<<<(E + 255) / 256, 256, 0, stream>>>(dst, offs, woff, csr, E);

    edge_logits<<<(E + 255) / 256, 256, 0, stream>>>(src, dst, eattr, c3, s1, s2, elog, E);

    aggregate<<<N, 256, 0, stream>>>(offs, csr, src, elog, xw, outh);

    // out = ELU(out_h @ proj_w + b)   (WMMA f32 + epilogue)
    gemm256_wmma<1><<<dim3(2, N / 16), 256, 0, stream>>>(outh, proj_w, proj_b, out);
}